// MelCepstralAnalysis_15032385536596
// MI455X (gfx1250) — compile-verified
//
#include <hip/hip_runtime.h>
#include <math.h>

// ---------------- problem constants ----------------
#define CEP_M     24
#define NC        25          // CEP_M + 1
#define FFT_N     1024
#define Hh        512
#define HP1       513
#define N_ITERS   4
#define ALPHA_F   0.35f
#define ROWS      16          // rows per workgroup (== WMMA M)
#define THREADS   256
#define FSTRIDE   1032        // FFT row stride (1024 + 8 dwords -> bank spread)
#define LOGSTRIDE 520
#define RT_LD     64
#define RT_PLANE  (ROWS * RT_LD)   // 1024 floats per partial plane
#define MC_LD     32
#define AUG_LD    27          // 25 matrix cols + rhs + solved-x slot
#define AUG_SYS   (NC * AUG_LD)

// packed fragment B matrices in d_ws: [tiles][32 lanes][2 floats]
// f1p: 129 k-tiles x 2 n-tiles   (A_FREQT,  scales 1/N and 0.5-endpoints folded)
// f2p: 33 n-tiles x 7 k-steps    (A_IFREQT)
// f3p: 129 k-tiles x 4 n-tiles   (A_RFREQT, scale 1/N folded)
#define F1P_SZ (129 * 2 * 64)
#define F2P_SZ (33 * 7 * 64)
#define F3P_SZ (129 * 4 * 64)

typedef __attribute__((ext_vector_type(2))) float v2f;
typedef __attribute__((ext_vector_type(8))) float v8f;

#if defined(__AMDGCN__) && __has_builtin(__builtin_amdgcn_wmma_f32_16x16x4_f32)
#define USE_WMMA 1
#else
#define USE_WMMA 0
#endif

static __device__ __forceinline__ v8f wmma_f32_4(v2f a, v2f b, v8f c) {
#if USE_WMMA
  // 8 args: (neg_a, A, neg_b, B, c_mod, C, reuse_a, reuse_b)
  return __builtin_amdgcn_wmma_f32_16x16x4_f32(false, a, false, b, (short)0, c,
                                               false, false);
#else
  c[0] += a.x * b.x;  // parse-only stub for host pass
  return c;
#endif
}

static __device__ __forceinline__ int br10(int n) {
  return (int)(__brev((unsigned)n) >> 22);
}

static __device__ __forceinline__ float ipow_signed(float base, int e) {
  float p = powf(fabsf(base), (float)e);
  if (base < 0.0f && (e & 1)) p = -p;
  return p;
}

// ---------------------------------------------------------------------------
// Setup kernel: generate the three frequency-transform matrices with the
// 2D recurrence A[i][j] = A[i-1][j-1] + alpha*(A[i][j-1] - A[i-1][j]),
// parallelized over anti-diagonals in LDS, then emitted pre-packed into
// WMMA B-fragment order (zero-padded, scales folded) so the main-kernel GEMM
// loops need no guards and no scale VALU.
// block 0: A_FREQT  (A:25x513, alpha=+0.35, freqt init)  -> f1p
// block 1: A_IFREQT (A:513x25, alpha=-0.35, freqt init)  -> f2p
// block 2: A_RFREQT (A:49x513, alpha=+0.35, coef  init)  -> f3p
// ---------------------------------------------------------------------------
__global__ __launch_bounds__(THREADS) void mel_setup_kernel(float* __restrict__ f1p,
                                                            float* __restrict__ f2p,
                                                            float* __restrict__ f3p) {
  extern __shared__ float A[];
  int L1, L2, mode;
  float alpha;
  if (blockIdx.x == 0)      { L1 = 513; L2 = 25;  alpha =  ALPHA_F; mode = 0; }
  else if (blockIdx.x == 1) { L1 = 25;  L2 = 513; alpha = -ALPHA_F; mode = 0; }
  else                      { L1 = 513; L2 = 49;  alpha =  ALPHA_F; mode = 1; }

  const int tid = threadIdx.x;
  const float invN = 1.0f / (float)FFT_N;
  // row 0
  for (int j = tid; j < L1; j += blockDim.x) {
    float p;
    if (mode == 0) p = ipow_signed(alpha, j);            // freqt: alpha^j
    else           p = (j == 0) ? 1.0f : 0.0f;           // coef:  delta
    A[j] = p;
  }
  // column 0, rows >= 1
  for (int i = 1 + tid; i < L2; i += blockDim.x) {
    A[i * L1] = (mode == 0) ? 0.0f : ipow_signed(-alpha, i);
  }
  __syncthreads();
  // anti-diagonal wavefront over i>=1, j>=1
  for (int d = 2; d <= (L1 - 1) + (L2 - 1); ++d) {
    int ilo = d - (L1 - 1); if (ilo < 1) ilo = 1;
    int ihi = d - 1;        if (ihi > L2 - 1) ihi = L2 - 1;
    for (int i = ilo + tid; i <= ihi; i += blockDim.x) {
      int j = d - i;
      A[i * L1 + j] = A[(i - 1) * L1 + (j - 1)] +
                      alpha * (A[i * L1 + (j - 1)] - A[(i - 1) * L1 + j]);
    }
    __syncthreads();
  }

  // emit packed B fragments (matrix element F[k][n] == A[n][k])
  if (blockIdx.x == 0) {
    for (int idx = tid; idx < F1P_SZ; idx += blockDim.x) {
      int v = idx & 1, lane = (idx >> 1) & 31;
      int nt = (idx >> 6) & 1, kt = idx >> 7;
      int k = 4 * kt + v + 2 * (lane >> 4);
      int n = nt * 16 + (lane & 15);
      float val = 0.0f;
      if (k <= Hh && n < NC)
        val = A[n * 513 + k] * invN * ((k == 0 || k == Hh) ? 0.5f : 1.0f);
      f1p[idx] = val;
    }
  } else if (blockIdx.x == 1) {
    for (int idx = tid; idx < F2P_SZ; idx += blockDim.x) {
      int v = idx & 1, lane = (idx >> 1) & 31;
      int g = idx >> 6;              // nt*7 + s
      int s = g % 7, nt = g / 7;
      int k = 4 * s + v + 2 * (lane >> 4);
      int n = nt * 16 + (lane & 15);
      f2p[idx] = (k < NC && n < HP1) ? A[n * 25 + k] : 0.0f;
    }
  } else {
    for (int idx = tid; idx < F3P_SZ; idx += blockDim.x) {
      int v = idx & 1, lane = (idx >> 1) & 31;
      int nt = (idx >> 6) & 3, kt = idx >> 8;
      int k = 4 * kt + v + 2 * (lane >> 4);
      int n = nt * 16 + (lane & 15);
      f3p[idx] = (k <= Hh && n < 49) ? A[n * 513 + k] * invN : 0.0f;
    }
  }
}

// ---------------------------------------------------------------------------
// 8 complex in-place radix-2 DIT FFTs of length 1024 in LDS, using a shared
// 512-entry {cos,sin} twiddle table (one ds_load_b64 per butterfly instead of
// sincos). Input pre-scattered in bit-reversed order. sign=-1 fwd, +1 inv
// (1/N scaling for the inverse is folded into the packed B matrices).
// ---------------------------------------------------------------------------
static __device__ void fft_block(float* fr, float* fi, const float* stw,
                                 float sign, int tid) {
#pragma unroll 1
  for (int s = 0; s < 10; ++s) {
    const int half = 1 << s;
    const int shift = 9 - s;
    for (int idx = tid; idx < 8 * 512; idx += THREADS) {
      const int row = idx >> 9;
      const int t = idx & 511;
      const int j = t & (half - 1);
      const int blk = t >> s;
      const int pos = row * FSTRIDE + blk * (half << 1) + j;
      const v2f w = *(const v2f*)(stw + 2 * (j << shift));
      const float wr = w.x;
      const float wi = sign * w.y;   // fwd: e^{-i0}, inv: e^{+i0}
      const float vr = fr[pos + half], vi = fi[pos + half];
      const float tr = vr * wr - vi * wi;
      const float ti = vr * wi + vi * wr;
      const float ur = fr[pos], ui = fi[pos];
      fr[pos] = ur + tr;
      fi[pos] = ui + ti;
      fr[pos + half] = ur - tr;
      fi[pos + half] = ui - ti;
    }
    __syncthreads();
  }
}

// ---------------------------------------------------------------------------
// Main kernel: 16 rows per block, 256 threads (8 wave32).
// ---------------------------------------------------------------------------
__global__ __launch_bounds__(THREADS) void mel_main_kernel(
    const float* __restrict__ x, float* __restrict__ out,
    const float* __restrict__ f1p, const float* __restrict__ f2p,
    const float* __restrict__ f3p) {
  extern __shared__ float sm[];
  float* fr   = sm;                         // 8*FSTRIDE  (paired FFT, real)
  float* fi   = fr + 8 * FSTRIDE;           // 8*FSTRIDE  (paired FFT, imag)
  float* slog = fi + 8 * FSTRIDE;           // 16*LOGSTRIDE : log x
  float* saug = slog + ROWS * LOGSTRIDE;    // 16*AUG_SYS : e-buffer / solver
  float* srt  = saug + ROWS * AUG_SYS;      // 2*RT_PLANE : rt partials / scratch
  float* smc  = srt + 2 * RT_PLANE;         // 16*MC_LD : mc (zero-padded)
  float* sav  = smc + ROWS * MC_LD;         // 32 : (-alpha)^i
  float* stw  = sav + 32;                   // 1024 : twiddle {cos,sin}(2pi t/N)

  const int tid  = threadIdx.x;
  const int wave = tid >> 5;
  const int lane = tid & 31;
  const int lo   = tid & 15;          // WMMA: M/N lane index
  const int hi   = (tid >> 4) & 1;    // WMMA: K/M-half selector
  const int koff = 2 * hi;
  const int row0 = blockIdx.x * ROWS;
  // A-fragment base: row m = lo of the 16-row tile lives at
  //   ((m & 1) ? fi : fr) + (m >> 1) * FSTRIDE   (branchless form below)
  const float* abase = fr + ((lo & 1) * 8 + (lo >> 1)) * FSTRIDE;

  if (tid == 0) {  // warm L2 path for the streamed packed-B matrices
    __builtin_prefetch(f2p, 0, 0);
    __builtin_prefetch(f3p, 0, 0);
  }
  if (tid < 32) sav[tid] = (tid < NC) ? ipow_signed(-ALPHA_F, tid) : 0.0f;

  // ---- twiddle table: stw[2t..2t+1] = {cos,sin}(2*pi*t/1024), t<512 ----
  for (int t = tid; t < 512; t += THREADS) {
    float sn, cs;
    __sincosf(0.006135923151542565f * (float)t, &sn, &cs);  // 2*pi/1024
    stw[2 * t] = cs;
    stw[2 * t + 1] = sn;
  }

  // ---- load x, take log ----
  for (int idx = tid; idx < ROWS * HP1; idx += THREADS) {
    int m = idx / HP1, k = idx - m * HP1;
    slog[m * LOGSTRIDE + k] = __logf(x[(size_t)(row0 + m) * HP1 + k]);
  }
  __syncthreads();

  // ---- scatter Hermitian-paired log-spectra (bit-reversed), inverse FFT ----
  for (int idx = tid; idx < 8 * 1024; idx += THREADS) {
    int p = idx >> 10, k = idx & 1023;
    int km = (k <= Hh) ? k : (FFT_N - k);
    int dst = p * FSTRIDE + br10(k);
    fr[dst] = slog[(2 * p) * LOGSTRIDE + km];
    fi[dst] = slog[(2 * p + 1) * LOGSTRIDE + km];
  }
  __syncthreads();
  fft_block(fr, fi, stw, +1.0f, tid);  // c = irfft(log x)

  // ---- GEMM1: mc = c[:,:513] @ F1 (M=16, K=516, N=32) ----
  // waves 0..3: (n-tile = wave&1, K-half = wave>>1); partials in srt scratch
  if (wave < 4) {
    const int nt = wave & 1, kh = wave >> 1;
    const v2f* bp = (const v2f*)f1p + nt * 32 + lane;  // + kt*64 per k-tile
    v8f acc0 = {}, acc1 = {};
    int kt = kh;
    for (; kt + 2 <= 128; kt += 4) {
      v2f a0 = *(const v2f*)(abase + 4 * kt + koff);
      v2f a1 = *(const v2f*)(abase + 4 * (kt + 2) + koff);
      acc0 = wmma_f32_4(a0, bp[kt * 64], acc0);
      acc1 = wmma_f32_4(a1, bp[(kt + 2) * 64], acc1);
    }
    if (kt <= 128) {  // tail (kt=128 for K-half 0; padded B rows zero > 512)
      v2f a = *(const v2f*)(abase + 4 * kt + koff);
      acc0 = wmma_f32_4(a, bp[kt * 64], acc0);
    }
    v8f acc = acc0 + acc1;
#pragma unroll
    for (int v = 0; v < 8; ++v)
      srt[kh * 512 + (v + 8 * hi) * MC_LD + nt * 16 + lo] = acc[v];
  }
  __syncthreads();
  for (int idx = tid; idx < ROWS * MC_LD; idx += THREADS)
    smc[idx] = srt[idx] + srt[512 + idx];
  __syncthreads();

  // ============================ Newton iterations ===========================
  for (int it = 0; it < N_ITERS; ++it) {
    // ---- GEMM2: c = mc @ F2 (K=28, N=528 -> 33 tiles), all 8 waves ----
    v2f a2[7];
#pragma unroll
    for (int s = 0; s < 7; ++s)
      a2[s] = *(const v2f*)(smc + lo * MC_LD + 4 * s + koff);
    v8f acc2[5];
#pragma unroll
    for (int tt = 0; tt < 5; ++tt) {
      const int t = wave + 8 * tt;                // wave-uniform tile index
      v8f acc = {};
      if (t < 33) {
        const v2f* bp = (const v2f*)f2p + t * 7 * 32 + lane;
#pragma unroll
        for (int s = 0; s < 7; ++s) acc = wmma_f32_4(a2[s], bp[s * 32], acc);
      }
      acc2[tt] = acc;
    }
    // zero-pad FFT buffers, scatter c at bit-reversed positions
    for (int idx = tid; idx < 8 * FSTRIDE; idx += THREADS) {
      fr[idx] = 0.0f;
      fi[idx] = 0.0f;
    }
    __syncthreads();
#pragma unroll
    for (int tt = 0; tt < 5; ++tt) {
      const int t = wave + 8 * tt;
      if (t < 33) {
        const int n = t * 16 + lo;
        if (n < HP1) {
          const int dst = br10(n);
#pragma unroll
          for (int v = 0; v < 8; ++v) {
            const int m = v + 8 * hi;
            float* bw = ((m & 1) ? fi : fr) + (m >> 1) * FSTRIDE;
            bw[dst] = acc2[tt][v];
          }
        }
      }
    }
    __syncthreads();
    fft_block(fr, fi, stw, -1.0f, tid);  // forward: rfft(c, 1024)

    // ---- d = Re(rfft(c)); e = exp(logx - 2 d) -> e-buffer (saug) ----
    for (int idx = tid; idx < ROWS * HP1; idx += THREADS) {
      int m = idx / HP1, k = idx - m * HP1;
      const float* bptr = ((m & 1) ? fi : fr) + (m >> 1) * FSTRIDE;
      int k2 = (FFT_N - k) & (FFT_N - 1);
      float dv = 0.5f * (bptr[k] + bptr[k2]);
      saug[m * HP1 + k] = __expf(slog[m * LOGSTRIDE + k] - 2.0f * dv);
    }
    __syncthreads();

    // ---- scatter Hermitian-paired e spectra, inverse FFT -> rd ----
    for (int idx = tid; idx < 8 * 1024; idx += THREADS) {
      int p = idx >> 10, k = idx & 1023;
      int km = (k <= Hh) ? k : (FFT_N - k);
      int dst = p * FSTRIDE + br10(k);
      fr[dst] = saug[(2 * p) * HP1 + km];
      fi[dst] = saug[(2 * p + 1) * HP1 + km];
    }
    __syncthreads();
    fft_block(fr, fi, stw, +1.0f, tid);  // rd = irfft(e)

    // ---- GEMM3: rt = rd[:,:513] @ F3 (N=64 -> 4 tiles), all 8 waves ----
    // (n-tile = wave&3, K-half = wave>>2); two partial planes in srt
    {
      const int nt = wave & 3, kh = wave >> 2;
      const v2f* bp = (const v2f*)f3p + nt * 32 + lane;  // + kt*128 per tile
      v8f acc0 = {}, acc1 = {};
      int kt = kh;
      for (; kt + 2 <= 128; kt += 4) {
        v2f a0 = *(const v2f*)(abase + 4 * kt + koff);
        v2f a1 = *(const v2f*)(abase + 4 * (kt + 2) + koff);
        acc0 = wmma_f32_4(a0, bp[kt * 128], acc0);
        acc1 = wmma_f32_4(a1, bp[(kt + 2) * 128], acc1);
      }
      if (kt <= 128) {
        v2f a = *(const v2f*)(abase + 4 * kt + koff);
        acc0 = wmma_f32_4(a, bp[kt * 128], acc0);
      }
      v8f acc = acc0 + acc1;
#pragma unroll
      for (int v = 0; v < 8; ++v)
        srt[kh * RT_PLANE + (v + 8 * hi) * RT_LD + nt * 16 + lo] = acc[v];
    }
    __syncthreads();

    // ---- build augmented systems (R + Q | ra), summing the two partials ----
    for (int idx = tid; idx < ROWS * NC * 26; idx += THREADS) {
      int m = idx / (NC * 26);
      int rem = idx - m * (NC * 26);
      int i = rem / 26, j = rem - i * 26;
      const float* r0 = srt + m * RT_LD;
      const float* r1 = r0 + RT_PLANE;
      float v;
      if (j < NC) {
        int dij = i - j; if (dij < 0) dij = -dij;
        v = (r0[dij] + r1[dij]) + (r0[i + j] + r1[i + j]);
      } else {
        v = (r0[i] + r1[i]) - sav[i];
      }
      saug[m * AUG_SYS + i * AUG_LD + j] = v;
    }
    __syncthreads();

    // ---- Gaussian elimination, 16 lanes per 25x25 system ----
    const int g = tid >> 4;
    const int sub = tid & 15;
    float* As = saug + g * AUG_SYS;
    for (int k = 0; k < NC; ++k) {
      const float pivinv = 1.0f / As[k * AUG_LD + k];
      for (int i = k + 1 + sub; i < NC; i += 16) {
        const float fmul = As[i * AUG_LD + k] * pivinv;
        for (int j = k; j < 26; ++j)
          As[i * AUG_LD + j] -= fmul * As[k * AUG_LD + j];
      }
      __syncthreads();
    }
    // ---- back substitution; mc += gradient ----
    for (int k = NC - 1; k >= 0; --k) {
      if (sub == 0) {
        const float xk = As[k * AUG_LD + 25] / As[k * AUG_LD + k];
        As[k * AUG_LD + 26] = xk;
        smc[g * MC_LD + k] += xk;
      }
      __syncthreads();
      const float xk = As[k * AUG_LD + 26];
      for (int i = sub; i < k; i += 16)
        As[i * AUG_LD + 25] -= As[i * AUG_LD + k] * xk;
      __syncthreads();
    }
  }

  // ---- output mc (16 x 25) ----
  for (int idx = tid; idx < ROWS * NC; idx += THREADS) {
    int m = idx / NC, j = idx - m * NC;
    out[(size_t)(row0 + m) * NC + j] = smc[m * MC_LD + j];
  }
}

// ---------------------------------------------------------------------------
extern "C" void kernel_launch(void* const* d_in, const int* in_sizes, int n_in,
                              void* d_out, int out_size, void* d_ws,
                              size_t ws_size, hipStream_t stream) {
  (void)n_in; (void)out_size; (void)ws_size;
  const float* x = (const float*)d_in[0];
  float* out = (float*)d_out;
  float* f1p = (float*)d_ws;
  float* f2p = f1p + F1P_SZ;
  float* f3p = f2p + F2P_SZ;

  const int nrows = in_sizes[0] / HP1;
  const int nblocks = nrows / ROWS;

  const size_t setup_lds = (size_t)(513 * 49) * sizeof(float);   // ~100.5 KB
  const size_t main_lds =
      (size_t)(2 * 8 * FSTRIDE + ROWS * LOGSTRIDE + ROWS * AUG_SYS +
               2 * RT_PLANE + ROWS * MC_LD + 32 + 1024) * sizeof(float);

  hipFuncSetAttribute(reinterpret_cast<const void*>(mel_setup_kernel),
                      hipFuncAttributeMaxDynamicSharedMemorySize,
                      (int)setup_lds);
  hipFuncSetAttribute(reinterpret_cast<const void*>(mel_main_kernel),
                      hipFuncAttributeMaxDynamicSharedMemorySize,
                      (int)main_lds);

  mel_setup_kernel<<<dim3(3), dim3(THREADS), setup_lds, stream>>>(f1p, f2p, f3p);
  mel_main_kernel<<<dim3(nblocks), dim3(THREADS), main_lds, stream>>>(x, out, f1p,
                                                                      f2p, f3p);
}